// OSSM_32384053412494
// MI455X (gfx1250) — compile-verified
//
#include <hip/hip_runtime.h>
#include <cstdint>

typedef float v2f __attribute__((ext_vector_type(2)));
typedef float v8f __attribute__((ext_vector_type(8)));

#define BB    4
#define HH    64
#define WWID  64
#define DMOD  96
#define DIN   192
#define KK    8
#define NST   16
#define RRK   6
#define LL    4096
#define CPROJ 38   // R + 2N
#define TL    64   // scan tile (steps staged per LDS buffer)

// ---------------------------------------------------------------------------
// scan order index math: position in (h,w) image for scan k at sequence idx l
// k: 0 row, 1 col, 2 row-flip, 3 col-flip, 4 diag, 5 antidiag, 6/7 flips
// ---------------------------------------------------------------------------
__device__ __forceinline__ int scan_pos(int k, int l) {
  int lf = ((k >> 1) & 1) ? (LL - 1 - l) : l;
  int mode = (k & 1) | ((k & 4) >> 1);   // 0:row 1:col 2:diag 3:antidiag
  int w = lf >> 6, h = lf & 63;
  switch (mode) {
    case 0:  return lf;
    case 1:  return (h << 6) | w;
    case 2:  return (h << 6) | ((h + w) & 63);
    default: return (h << 6) | ((w - h) & 63);
  }
}

// CDNA5 async copy: global -> LDS, 16B per lane, tracked by ASYNCcnt.
__device__ __forceinline__ void async_g2l_b128(const float* gptr, float* lptr) {
  uint32_t laddr = (uint32_t)(uintptr_t)lptr;   // low 32 bits = LDS offset
  uint64_t gaddr = (uint64_t)(uintptr_t)gptr;
  asm volatile("global_load_async_to_lds_b128 %0, %1, off"
               :: "v"(laddr), "v"(gaddr) : "memory");
}
__device__ __forceinline__ void wait_asynccnt0() {
  asm volatile("s_wait_asynccnt 0x0" ::: "memory");
}

// ---------------------------------------------------------------------------
// Generic C = A(MxK, row major) * W^T  with W given as (N x K) row major.
// One wave per 16x16 output tile, V_WMMA_F32_16X16X4_F32 over the K dim.
// Grid must cover exactly (M/16)*(N/16) waves.
// ---------------------------------------------------------------------------
__global__ void wmma_gemm_nt(const float* __restrict__ A,
                             const float* __restrict__ Wt,
                             float* __restrict__ C,
                             int M, int Nn, int Kd) {
  int wave = (blockIdx.x * blockDim.x + threadIdx.x) >> 5;
  int lane = threadIdx.x & 31;
  int ntiles = Nn >> 4;
  int tm = wave / ntiles;
  int tn = wave - tm * ntiles;
  if (tm >= (M >> 4)) return;             // wave-uniform guard
  int lane15 = lane & 15;
  int khalf  = (lane >> 4) << 1;          // lanes 0-15: K 0,1  lanes 16-31: K 2,3
  int row = (tm << 4) + lane15;
  int col = (tn << 4) + lane15;
  const float* Ar = A  + (size_t)row * Kd + khalf;
  const float* Wr = Wt + (size_t)col * Kd + khalf;
  v8f acc = {0.f, 0.f, 0.f, 0.f, 0.f, 0.f, 0.f, 0.f};
  for (int k0 = 0; k0 < Kd; k0 += 4) {
    v2f a = *(const v2f*)(Ar + k0);
    v2f b = *(const v2f*)(Wr + k0);
    acc = __builtin_amdgcn_wmma_f32_16x16x4_f32(false, a, false, b,
                                                (short)0, acc, false, false);
  }
  int mbase = (tm << 4) + ((lane >> 4) << 3);   // M = v + 8*(lane>=16)
#pragma unroll
  for (int v = 0; v < 8; ++v)
    C[(size_t)(mbase + v) * Nn + col] = acc[v];
}

// ---------------------------------------------------------------------------
// Depthwise 3x3 SAME conv + bias + SiLU. Reads xc from the packed xz buffer
// (channels-last, first DIN of 2*DIN), writes xconv in (B,DIN,H,W).
// ---------------------------------------------------------------------------
__global__ void conv_silu_kernel(const float* __restrict__ xz,
                                 const float* __restrict__ cw,
                                 const float* __restrict__ cb,
                                 float* __restrict__ xconv) {
  int id = blockIdx.x * blockDim.x + threadIdx.x;
  if (id >= BB * DIN * LL) return;
  int d = id % DIN;
  int t = id / DIN;
  int p = t & (LL - 1);
  int b = t >> 12;
  int h = p >> 6, w = p & 63;
  float acc = cb[d];
  const float* wk = cw + d * 9;
#pragma unroll
  for (int kh = 0; kh < 3; ++kh) {
    int hh = h + kh - 1;
    if ((unsigned)hh >= (unsigned)HH) continue;
#pragma unroll
    for (int kw = 0; kw < 3; ++kw) {
      int wi = w + kw - 1;
      if ((unsigned)wi >= (unsigned)WWID) continue;
      acc += wk[kh * 3 + kw] *
             xz[(size_t)(((b * HH + hh) << 6) + wi) * (2 * DIN) + d];
    }
  }
  acc = acc / (1.f + __expf(-acc));      // SiLU
  xconv[(size_t)(b * DIN + d) * LL + p] = acc;
}

// ---------------------------------------------------------------------------
// x_dbl[b,k,c,l] = sum_d xpw[k,c,d] * xconv[b,d,pos(k,l)]  (cross-scan fused
// as an on-the-fly gather).  WMMA over d, 16x16 output tiles, rows padded
// 38 -> 48 with a value mask (keeps EXEC all-1s for WMMA).
// ---------------------------------------------------------------------------
__global__ void xproj_wmma_kernel(const float* __restrict__ xconv,
                                  const float* __restrict__ xpw,
                                  float* __restrict__ xdbl) {
  int wave = (blockIdx.x * blockDim.x + threadIdx.x) >> 5;
  int lane = threadIdx.x & 31;
  int lt = wave & 255; wave >>= 8;       // 256 L-tiles
  int ct = wave % 3;   wave /= 3;        // 3 C-tiles (48 rows, 38 valid)
  int k  = wave & 7;
  int b  = wave >> 3;
  if (b >= BB) return;
  int lane15 = lane & 15;
  int khalf  = (lane >> 4) << 1;
  int c  = (ct << 4) + lane15;
  int cc = c < CPROJ ? c : (CPROJ - 1);
  float msk = c < CPROJ ? 1.f : 0.f;
  int l = (lt << 4) + lane15;
  int p = scan_pos(k, l);
  const float* Wk = xpw + (size_t)k * CPROJ * DIN + (size_t)cc * DIN + khalf;
  const float* xb = xconv + (size_t)b * DIN * LL + (size_t)khalf * LL + p;
  v8f acc = {0.f, 0.f, 0.f, 0.f, 0.f, 0.f, 0.f, 0.f};
  for (int kk = 0; kk < DIN; kk += 4) {
    v2f a = *(const v2f*)(Wk + kk);
    a.x *= msk; a.y *= msk;
    v2f bv;
    bv.x = xb[(size_t)kk * LL];
    bv.y = xb[(size_t)(kk + 1) * LL];
    acc = __builtin_amdgcn_wmma_f32_16x16x4_f32(false, a, false, bv,
                                                (short)0, acc, false, false);
  }
  float* outb = xdbl + (size_t)(b * KK + k) * CPROJ * LL;
  int mb = (ct << 4) + ((lane >> 4) << 3);
#pragma unroll
  for (int v = 0; v < 8; ++v) {
    int m = mb + v;
    if (m < CPROJ) outb[(size_t)m * LL + l] = acc[v];
  }
}

// ---------------------------------------------------------------------------
// Selective scan. One block per (b,k) = 192 threads (one per d), 6 waves.
// Fuses: dt = softplus(dt_w @ dt_low + bias), state recurrence over N=16,
// y += D*u.  The 38 shared x_dbl rows are staged through a double-buffered
// LDS tile via CDNA5 async global->LDS copies (ASYNCcnt), so HBM latency of
// tile t+1 hides behind the 64 serial recurrence steps of tile t.
// ---------------------------------------------------------------------------
__global__ void __launch_bounds__(DIN)
scan_kernel(const float* __restrict__ xconv,
            const float* __restrict__ xdbl,
            const float* __restrict__ dtw_all,
            const float* __restrict__ dtb,
            const float* __restrict__ A_logs,
            const float* __restrict__ Ds,
            float* __restrict__ ys) {
  __shared__ __align__(16) float sd[2][CPROJ * TL];   // 2 x 9728 B
  int bk = blockIdx.x;
  int b = bk >> 3, k = bk & 7;
  int d = threadIdx.x;
  int kd = k * DIN + d;
  float dtw[RRK];
#pragma unroll
  for (int r = 0; r < RRK; ++r) dtw[r] = dtw_all[(size_t)kd * RRK + r];
  float bias = dtb[kd];
  float a[NST];
#pragma unroll
  for (int n = 0; n < NST; ++n) a[n] = -__expf(A_logs[(size_t)kd * NST + n]);
  float Dv = Ds[kd];
  float hreg[NST];
#pragma unroll
  for (int n = 0; n < NST; ++n) hreg[n] = 0.f;
  const float* xcb = xconv + (size_t)(b * DIN + d) * LL;
  const float* xdb = xdbl + (size_t)(b * KK + k) * CPROJ * LL;
  float* ysb = ys + (size_t)((b * KK + k) * DIN + d) * LL;

  const int NCHUNK = CPROJ * (TL / 4);   // 608 16B chunks per tile
  // prologue: async-stage tile 0 into buffer 0
  for (int idx = threadIdx.x; idx < NCHUNK; idx += DIN) {
    int c = idx >> 4;               // row (0..37)
    int i4 = (idx & 15) << 2;       // column group
    async_g2l_b128(xdb + (size_t)c * LL + i4, &sd[0][c * TL + i4]);
  }

  for (int t = 0; t < LL / TL; ++t) {
    int l0 = t * TL;
    const float* cur = sd[t & 1];
    wait_asynccnt0();               // own async loads for this tile done
    __syncthreads();                // -> all waves' loads done; prev reads done
    if (l0 + TL < LL) {             // stage next tile while computing this one
      float* nxt = sd[(t + 1) & 1];
      for (int idx = threadIdx.x; idx < NCHUNK; idx += DIN) {
        int c = idx >> 4;
        int i4 = (idx & 15) << 2;
        async_g2l_b128(xdb + (size_t)c * LL + (l0 + TL) + i4,
                       &nxt[c * TL + i4]);
      }
    }
    for (int i0 = 0; i0 < TL; i0 += 8) {
      float ubuf[8];                // pipeline the 8 u-gathers (clause-able)
#pragma unroll
      for (int j = 0; j < 8; ++j)
        ubuf[j] = xcb[scan_pos(k, l0 + i0 + j)];
#pragma unroll
      for (int j = 0; j < 8; ++j) {
        int i = i0 + j;
        float xr = bias;
#pragma unroll
        for (int r = 0; r < RRK; ++r) xr += dtw[r] * cur[r * TL + i];
        float dt = (xr > 20.f) ? xr : log1pf(__expf(xr));  // softplus
        float u  = ubuf[j];
        float du = dt * u;
        float y  = 0.f;
#pragma unroll
        for (int n = 0; n < NST; ++n) {
          hreg[n] = __expf(dt * a[n]) * hreg[n] + du * cur[(RRK + n) * TL + i];
          y += hreg[n] * cur[(RRK + NST + n) * TL + i];
        }
        ysb[l0 + i] = y + Dv * u;
      }
    }
  }
}

// ---------------------------------------------------------------------------
// Cross-merge: y[b,p,d] = sum_k ys[b,k,d, inv_l(k,p)] (deterministic gather).
// Output layout (B, L, DIN) ready for LayerNorm.
// ---------------------------------------------------------------------------
__global__ void merge_kernel(const float* __restrict__ ys,
                             float* __restrict__ y) {
  int id = blockIdx.x * blockDim.x + threadIdx.x;
  if (id >= BB * LL * DIN) return;
  int d = id % DIN;
  int t = id / DIN;
  int p = t & (LL - 1);
  int b = t >> 12;
  int h = p >> 6, w = p & 63;
  int l0 = (h << 6) | w;
  int l1 = (w << 6) | h;
  int l4 = (((w - h) & 63) << 6) | h;
  int l5 = (((w + h) & 63) << 6) | h;
  const size_t CHs = (size_t)DIN * LL;
  const float* base = ys + ((size_t)b * KK * DIN + d) * LL;
  float s = base[l0]
          + base[CHs + l1]
          + base[2 * CHs + (LL - 1 - l0)]
          + base[3 * CHs + (LL - 1 - l1)]
          + base[4 * CHs + l4]
          + base[5 * CHs + l5]
          + base[6 * CHs + (LL - 1 - l4)]
          + base[7 * CHs + (LL - 1 - l5)];
  y[id] = s;
}

// ---------------------------------------------------------------------------
// LayerNorm over DIN + SiLU(z) gate.  One thread per token row.
// ---------------------------------------------------------------------------
__global__ void ln_silu_kernel(const float* __restrict__ y,
                               const float* __restrict__ xz,
                               const float* __restrict__ lnw,
                               const float* __restrict__ lnb,
                               float* __restrict__ yln) {
  int row = blockIdx.x * blockDim.x + threadIdx.x;
  if (row >= BB * LL) return;
  const float* yr = y + (size_t)row * DIN;
  float s = 0.f, s2 = 0.f;
  for (int dd = 0; dd < DIN; ++dd) { float v = yr[dd]; s += v; s2 += v * v; }
  float mu  = s * (1.f / DIN);
  float var = s2 * (1.f / DIN) - mu * mu;
  float inv = rsqrtf(var + 1e-5f);
  const float* zr = xz + (size_t)row * (2 * DIN) + DIN;
  float* orow = yln + (size_t)row * DIN;
  for (int dd = 0; dd < DIN; ++dd) {
    float v = (yr[dd] - mu) * inv * lnw[dd] + lnb[dd];
    float z = zr[dd];
    orow[dd] = v * (z / (1.f + __expf(-z)));
  }
}

// ---------------------------------------------------------------------------
extern "C" void kernel_launch(void* const* d_in, const int* in_sizes, int n_in,
                              void* d_out, int out_size, void* d_ws, size_t ws_size,
                              hipStream_t stream) {
  (void)in_sizes; (void)n_in; (void)out_size; (void)ws_size;
  const float* x    = (const float*)d_in[0];
  const float* ipw  = (const float*)d_in[1];
  const float* cw   = (const float*)d_in[2];
  const float* cb   = (const float*)d_in[3];
  const float* xpw  = (const float*)d_in[4];
  const float* dtw  = (const float*)d_in[5];
  const float* dtb  = (const float*)d_in[6];
  const float* alog = (const float*)d_in[7];
  const float* Ds   = (const float*)d_in[8];
  const float* lnw  = (const float*)d_in[9];
  const float* lnb  = (const float*)d_in[10];
  const float* opw  = (const float*)d_in[11];
  float* out = (float*)d_out;

  float* ws    = (float*)d_ws;
  float* xz    = ws;                                   // 16384 * 384
  float* xconv = xz    + (size_t)BB * LL * 2 * DIN;    // 4*192*4096
  float* xdbl  = xconv + (size_t)BB * DIN * LL;        // 4*8*38*4096
  float* ysb   = xdbl  + (size_t)BB * KK * CPROJ * LL; // 4*8*192*4096
  float* ymg   = ysb   + (size_t)BB * KK * DIN * LL;   // 16384*192
  float* yln   = ymg   + (size_t)BB * LL * DIN;        // 16384*192

  // 1. in_proj: (16384 x 96) @ (384 x 96)^T -> xz       1024*24 tiles
  wmma_gemm_nt<<<3072, 256, 0, stream>>>(x, ipw, xz, BB * LL, 2 * DIN, DMOD);
  // 2. depthwise conv + SiLU
  conv_silu_kernel<<<(BB * DIN * LL) / 256, 256, 0, stream>>>(xz, cw, cb, xconv);
  // 3. x_dbl projections (cross-scan fused gather)      4*8*3*256 tiles
  xproj_wmma_kernel<<<3072, 256, 0, stream>>>(xconv, xpw, xdbl);
  // 4. selective scan (fused dt-proj + softplus + D*u, async double-buffer)
  scan_kernel<<<BB * KK, DIN, 0, stream>>>(xconv, xdbl, dtw, dtb, alog, Ds, ysb);
  // 5. cross-merge
  merge_kernel<<<(BB * LL * DIN) / 256, 256, 0, stream>>>(ysb, ymg);
  // 6. LayerNorm + SiLU gate
  ln_silu_kernel<<<(BB * LL) / 256, 256, 0, stream>>>(ymg, xz, lnw, lnb, yln);
  // 7. out_proj: (16384 x 192) @ (96 x 192)^T -> out    1024*6 tiles
  wmma_gemm_nt<<<768, 256, 0, stream>>>(yln, opw, out, BB * LL, DMOD, DIN);
}